// FeaturePropagation_19816979104412
// MI455X (gfx1250) — compile-verified
//
#include <hip/hip_runtime.h>
#include <math.h>

typedef __attribute__((ext_vector_type(2))) float v2f;
typedef __attribute__((ext_vector_type(8))) float v8f;

#define B_   16
#define N_   4096
#define S_   1024
#define DT_  128
#define DS_  256
#define CIN_ 384
#define C1_  256
#define C2_  256
#define M_   (B_ * N_)   // 65536

// ---- workspace layout (in floats) ----
#define OFF_X     ((size_t)0)                         // 65536*384
#define OFF_Y1    (OFF_X + (size_t)M_ * CIN_)         // 65536*256
#define OFF_IDX   (OFF_Y1 + (size_t)M_ * C1_)         // 65536*3 ints
#define OFF_WGT   (OFF_IDX + (size_t)M_ * 3)          // 65536*3 floats
#define OFF_STATS (OFF_WGT + (size_t)M_ * 3)          // 8*256 floats
// stats sublayout: [sum1|sq1|sum2|sq2|sc1|sh1|sc2|sh2], 256 each

__global__ void zero_stats_kernel(float* __restrict__ stats) {
    stats[threadIdx.x] = 0.0f;   // 1024 threads cover sum1,sq1,sum2,sq2
}

// ---- K1: 3-NN + inverse-distance weights --------------------------------
__global__ void knn_kernel(const float* __restrict__ xyzT,
                           const float* __restrict__ xyzS,
                           int*   __restrict__ idx_out,
                           float* __restrict__ w_out) {
    __shared__ float sx[S_], sy[S_], sz[S_], sn[S_];
    const int b = blockIdx.y;
    const float* src = xyzS + (size_t)b * S_ * 3;
    for (int i = threadIdx.x; i < S_; i += blockDim.x) {
        float x = src[i * 3 + 0], y = src[i * 3 + 1], z = src[i * 3 + 2];
        sx[i] = x; sy[i] = y; sz[i] = z; sn[i] = x * x + y * y + z * z;
    }
    __syncthreads();

    const int n = blockIdx.x * blockDim.x + threadIdx.x;
    const float* t = xyzT + ((size_t)b * N_ + n) * 3;
    const float tx = t[0], ty = t[1], tz = t[2];
    const float tn = tx * tx + ty * ty + tz * tz;

    float d0 = 3.4e38f, d1 = 3.4e38f, d2 = 3.4e38f;
    int   i0 = 0, i1 = 0, i2 = 0;
    for (int s = 0; s < S_; ++s) {
        float dot = tx * sx[s] + ty * sy[s] + tz * sz[s];
        float d = fmaxf(tn + sn[s] - 2.0f * dot, 0.0f);
        if (d < d0)      { d2 = d1; i2 = i1; d1 = d0; i1 = i0; d0 = d; i0 = s; }
        else if (d < d1) { d2 = d1; i2 = i1; d1 = d;  i1 = s; }
        else if (d < d2) { d2 = d;  i2 = s; }
    }
    float r0 = 1.0f / (d0 + 1e-8f);
    float r1 = 1.0f / (d1 + 1e-8f);
    float r2 = 1.0f / (d2 + 1e-8f);
    float rs = 1.0f / (r0 + r1 + r2);

    const size_t p = (size_t)b * N_ + n;
    idx_out[p * 3 + 0] = i0; idx_out[p * 3 + 1] = i1; idx_out[p * 3 + 2] = i2;
    w_out[p * 3 + 0] = r0 * rs; w_out[p * 3 + 1] = r1 * rs; w_out[p * 3 + 2] = r2 * rs;
}

// ---- K2: interpolate + concat (wave32 per point) ------------------------
__global__ void interp_concat_kernel(const float* __restrict__ ft,
                                     const float* __restrict__ fs,
                                     const int*   __restrict__ idx,
                                     const float* __restrict__ w,
                                     float* __restrict__ x) {
    const int gid  = blockIdx.x * blockDim.x + threadIdx.x;
    const int p    = gid >> 5;
    const int lane = gid & 31;
    const int b    = p >> 12;      // p / 4096

    const int   j0 = idx[(size_t)p * 3 + 0];
    const int   j1 = idx[(size_t)p * 3 + 1];
    const int   j2 = idx[(size_t)p * 3 + 2];
    const float w0 = w[(size_t)p * 3 + 0];
    const float w1 = w[(size_t)p * 3 + 1];
    const float w2 = w[(size_t)p * 3 + 2];

    const float* f0 = fs + ((size_t)b * S_ + j0) * DS_;
    const float* f1 = fs + ((size_t)b * S_ + j1) * DS_;
    const float* f2 = fs + ((size_t)b * S_ + j2) * DS_;
    const float* tp = ft + (size_t)p * DT_;
    float*       xp = x  + (size_t)p * CIN_;

#pragma unroll
    for (int ch = 0; ch < DT_ / 32; ++ch) {
        int c = ch * 32 + lane;
        xp[c] = tp[c];
    }
#pragma unroll
    for (int ch = 0; ch < DS_ / 32; ++ch) {
        int c = ch * 32 + lane;
        xp[DT_ + c] = w0 * f0[c] + w1 * f1[c] + w2 * f2[c];
    }
}

// ---- K3: GEMM1 (x[M,384] @ W1^T -> y1[M,256]) + BN1 partial stats -------
// One wave computes a 16x64 output tile: 4 accumulators share each A-load.
// Uniform scalar K-loop (trip count compile-time constant) so the compiler
// can unroll and software-pipeline the loads.
__global__ void gemm1_kernel(const float* __restrict__ x,
                             const float* __restrict__ W1,
                             float* __restrict__ y1,
                             float* __restrict__ ssum,
                             float* __restrict__ ssq) {
    const int lane = threadIdx.x & 31;
    const int wave = threadIdx.x >> 5;
    const int m0   = (blockIdx.x * 8 + wave) * 16;
    const int n0   = blockIdx.y * 64;
    const int half = lane >> 4;     // 0: K=k,k+1   1: K=k+2,k+3
    const int lan  = lane & 15;

    const float* arow = x + (size_t)(m0 + lan) * CIN_ + 2 * half;
    const float* br0  = W1 + (size_t)(n0 +  0 + lan) * CIN_ + 2 * half;
    const float* br1  = W1 + (size_t)(n0 + 16 + lan) * CIN_ + 2 * half;
    const float* br2  = W1 + (size_t)(n0 + 32 + lan) * CIN_ + 2 * half;
    const float* br3  = W1 + (size_t)(n0 + 48 + lan) * CIN_ + 2 * half;

    v8f c0 = {}, c1 = {}, c2 = {}, c3 = {};
#pragma unroll 4
    for (int kk = 0; kk < CIN_ / 4; ++kk) {
        const int k = kk * 4;
        v2f a  = *(const v2f*)(arow + k);
        v2f q0 = *(const v2f*)(br0 + k);
        v2f q1 = *(const v2f*)(br1 + k);
        v2f q2 = *(const v2f*)(br2 + k);
        v2f q3 = *(const v2f*)(br3 + k);
        c0 = __builtin_amdgcn_wmma_f32_16x16x4_f32(false, a, false, q0, (short)0, c0, false, false);
        c1 = __builtin_amdgcn_wmma_f32_16x16x4_f32(false, a, false, q1, (short)0, c1, false, false);
        c2 = __builtin_amdgcn_wmma_f32_16x16x4_f32(false, a, false, q2, (short)0, c2, false, false);
        c3 = __builtin_amdgcn_wmma_f32_16x16x4_f32(false, a, false, q3, (short)0, c3, false, false);
    }

    float* ybase = y1 + (size_t)(m0 + half * 8) * C1_ + lan;
#pragma unroll
    for (int j = 0; j < 4; ++j) {
        v8f c = (j == 0) ? c0 : (j == 1) ? c1 : (j == 2) ? c2 : c3;
        float* yb = ybase + n0 + j * 16;
        float s = 0.0f, s2 = 0.0f;
#pragma unroll
        for (int r = 0; r < 8; ++r) {
            float v = c[r];
            yb[(size_t)r * C1_] = v;
            s += v; s2 += v * v;
        }
        atomicAdd(&ssum[n0 + j * 16 + lan], s);
        atomicAdd(&ssq[n0 + j * 16 + lan], s2);
    }
}

// ---- K5: GEMM2 with fused BN1+ReLU on A-load, writes raw y2 to out ------
__global__ void gemm2_kernel(const float* __restrict__ y1,
                             const float* __restrict__ W2,
                             const float* __restrict__ sc,
                             const float* __restrict__ sh,
                             float* __restrict__ y2,
                             float* __restrict__ ssum,
                             float* __restrict__ ssq) {
    const int lane = threadIdx.x & 31;
    const int wave = threadIdx.x >> 5;
    const int m0   = (blockIdx.x * 8 + wave) * 16;
    const int n0   = blockIdx.y * 64;
    const int half = lane >> 4;
    const int lan  = lane & 15;

    const float* arow = y1 + (size_t)(m0 + lan) * C1_ + 2 * half;
    const float* scp  = sc + 2 * half;
    const float* shp  = sh + 2 * half;
    const float* br0  = W2 + (size_t)(n0 +  0 + lan) * C1_ + 2 * half;
    const float* br1  = W2 + (size_t)(n0 + 16 + lan) * C1_ + 2 * half;
    const float* br2  = W2 + (size_t)(n0 + 32 + lan) * C1_ + 2 * half;
    const float* br3  = W2 + (size_t)(n0 + 48 + lan) * C1_ + 2 * half;

    v8f c0 = {}, c1 = {}, c2 = {}, c3 = {};
#pragma unroll 4
    for (int kk = 0; kk < C1_ / 4; ++kk) {
        const int k = kk * 4;
        v2f raw = *(const v2f*)(arow + k);
        v2f a;
        a.x = fmaxf(raw.x * scp[k]     + shp[k],     0.0f);
        a.y = fmaxf(raw.y * scp[k + 1] + shp[k + 1], 0.0f);
        v2f q0 = *(const v2f*)(br0 + k);
        v2f q1 = *(const v2f*)(br1 + k);
        v2f q2 = *(const v2f*)(br2 + k);
        v2f q3 = *(const v2f*)(br3 + k);
        c0 = __builtin_amdgcn_wmma_f32_16x16x4_f32(false, a, false, q0, (short)0, c0, false, false);
        c1 = __builtin_amdgcn_wmma_f32_16x16x4_f32(false, a, false, q1, (short)0, c1, false, false);
        c2 = __builtin_amdgcn_wmma_f32_16x16x4_f32(false, a, false, q2, (short)0, c2, false, false);
        c3 = __builtin_amdgcn_wmma_f32_16x16x4_f32(false, a, false, q3, (short)0, c3, false, false);
    }

    float* ybase = y2 + (size_t)(m0 + half * 8) * C2_ + lan;
#pragma unroll
    for (int j = 0; j < 4; ++j) {
        v8f c = (j == 0) ? c0 : (j == 1) ? c1 : (j == 2) ? c2 : c3;
        float* yb = ybase + n0 + j * 16;
        float s = 0.0f, s2 = 0.0f;
#pragma unroll
        for (int r = 0; r < 8; ++r) {
            float v = c[r];
            yb[(size_t)r * C2_] = v;
            s += v; s2 += v * v;
        }
        atomicAdd(&ssum[n0 + j * 16 + lan], s);
        atomicAdd(&ssq[n0 + j * 16 + lan], s2);
    }
}

// ---- K4/K6: finalize BN affine ------------------------------------------
__global__ void bn_finalize_kernel(const float* __restrict__ ssum,
                                   const float* __restrict__ ssq,
                                   const float* __restrict__ gamma,
                                   const float* __restrict__ beta,
                                   float* __restrict__ sc,
                                   float* __restrict__ sh) {
    const int n = threadIdx.x;
    const float inv_m = 1.0f / (float)M_;
    float mean = ssum[n] * inv_m;
    float var  = fmaxf(ssq[n] * inv_m - mean * mean, 0.0f);
    float s    = rsqrtf(var + 1e-5f) * gamma[n];
    sc[n] = s;
    sh[n] = beta[n] - mean * s;
}

// ---- K7: in-place BN2 + ReLU on output ----------------------------------
__global__ void apply_bn_relu_kernel(float* __restrict__ out,
                                     const float* __restrict__ sc,
                                     const float* __restrict__ sh) {
    const size_t i = (size_t)blockIdx.x * blockDim.x + threadIdx.x;
    const int n = (int)(i & (C2_ - 1));
    out[i] = fmaxf(out[i] * sc[n] + sh[n], 0.0f);
}

extern "C" void kernel_launch(void* const* d_in, const int* in_sizes, int n_in,
                              void* d_out, int out_size, void* d_ws, size_t ws_size,
                              hipStream_t stream) {
    const float* xyzT = (const float*)d_in[0];
    const float* xyzS = (const float*)d_in[1];
    const float* ft   = (const float*)d_in[2];
    const float* fs   = (const float*)d_in[3];
    const float* W1   = (const float*)d_in[4];
    const float* g1   = (const float*)d_in[5];
    const float* b1   = (const float*)d_in[6];
    const float* W2   = (const float*)d_in[7];
    const float* g2   = (const float*)d_in[8];
    const float* b2   = (const float*)d_in[9];
    float* out = (float*)d_out;

    float* ws    = (float*)d_ws;
    float* x     = ws + OFF_X;
    float* y1    = ws + OFF_Y1;
    int*   idx   = (int*)(ws + OFF_IDX);
    float* wgt   = ws + OFF_WGT;
    float* stats = ws + OFF_STATS;
    float* sum1 = stats + 0 * 256, *sq1 = stats + 1 * 256;
    float* sum2 = stats + 2 * 256, *sq2 = stats + 3 * 256;
    float* sc1  = stats + 4 * 256, *sh1 = stats + 5 * 256;
    float* sc2  = stats + 6 * 256, *sh2 = stats + 7 * 256;

    zero_stats_kernel<<<1, 1024, 0, stream>>>(stats);

    knn_kernel<<<dim3(N_ / 256, B_), 256, 0, stream>>>(xyzT, xyzS, idx, wgt);

    interp_concat_kernel<<<(M_ * 32) / 256, 256, 0, stream>>>(ft, fs, idx, wgt, x);

    gemm1_kernel<<<dim3(M_ / 128, C1_ / 64), 256, 0, stream>>>(x, W1, y1, sum1, sq1);

    bn_finalize_kernel<<<1, 256, 0, stream>>>(sum1, sq1, g1, b1, sc1, sh1);

    gemm2_kernel<<<dim3(M_ / 128, C2_ / 64), 256, 0, stream>>>(y1, W2, sc1, sh1, out, sum2, sq2);

    bn_finalize_kernel<<<1, 256, 0, stream>>>(sum2, sq2, g2, b2, sc2, sh2);

    apply_bn_relu_kernel<<<(M_ * C2_) / 256, 256, 0, stream>>>(out, sc2, sh2);
}